// W2Loss_68745246539870
// MI455X (gfx1250) — compile-verified
//
#include <hip/hip_runtime.h>
#include <hip/hip_bf16.h>

// ---------------------------------------------------------------------------
// W2 transport loss, gfx1250 (CDNA5, wave32, WMMA + TDM)
//
// Sizes (from reference): B=2048, T=4097, P=4097, down=16
//   -> 257 spline knots, 256 intervals, constant tridiagonal matrix A.
// CDNA5-specific paths:
//   * Batched tridiagonal solve recast as M = RHS x Ainv (2048x256x256 GEMM)
//     on V_WMMA_F32_16X16X4_F32 (f32 accumulate, exact-precision).
//   * Per-row 16 KB input tiles staged into LDS with the Tensor Data Mover
//     (tensor_load_to_lds + s_wait_tensorcnt), overlapping DMA with the
//     spline-coefficient setup work.
// ---------------------------------------------------------------------------

typedef __attribute__((ext_vector_type(2))) float        v2f;
typedef __attribute__((ext_vector_type(8))) float        v8f;
typedef __attribute__((ext_vector_type(4))) unsigned int u32x4;
typedef __attribute__((ext_vector_type(8))) int          i32x8;
typedef __attribute__((ext_vector_type(4))) int          i32x4;

#define TLEN 4097      // time samples
#define NK   257       // spline knots
#define NI   256       // spline intervals (padded m-vector length)
#define BLK  256       // block size for per-row kernels (8 wave32)
#define EPSF 1e-3f

// ---------------------------------------------------------------------------
// TDM: DMA one row of `nelem` f32 from global memory into LDS at byte offset
// `lds_byte_off`.  D# built per CDNA5 ISA section 8.3/8.4 (1-D tile, 4-byte
// elements, no padding / iterate / multicast).  Caller must be wave-uniform.
// This toolchain exposes the 6-arg builtin:
//   (u32x4 g0, i32x8 g1, i32x4 g2, i32x4 g3, i32x8 extra, i32 cpol)
// ---------------------------------------------------------------------------
__device__ __forceinline__ void tdm_load_row_f32(unsigned lds_byte_off,
                                                 const float* gsrc, int nelem)
{
    unsigned long long ga = (unsigned long long)(const void*)gsrc;  // byte addr
    u32x4 g0;
    g0[0] = 1u;                                          // count=1, user D#
    g0[1] = lds_byte_off;                                // lds_addr
    g0[2] = (unsigned)(ga & 0xFFFFFFFFull);              // global_addr[31:0]
    g0[3] = (unsigned)((ga >> 32) & 0x01FFFFFFull)       // global_addr[56:32]
          | 0x80000000u;                                 // type = 2 ("image")
    i32x8 g1;
    unsigned n = (unsigned)nelem;
    g1[0] = 0x00020000;                                  // data_size=2 (4B)
    g1[1] = (int)((n & 0xFFFFu) << 16);                  // tensor_dim0[15:0]
    g1[2] = (int)(((n >> 16) & 0xFFFFu) | (1u << 16));   // dim0[31:16] | tensor_dim1=1
    g1[3] = (int)((n & 0xFFFFu) << 16);                  // dim1[31:16]=0 | tile_dim0
    g1[4] = 0;                                           // tile_dim1/2 = 0 (unused)
    g1[5] = (int)n;                                      // tensor_dim0_stride[31:0]
    g1[6] = 0;                                           // stride hi | dim1_stride lo
    g1[7] = 0;                                           // dim1_stride hi
    i32x4 z4 = {0, 0, 0, 0};                             // groups 2/3 unused (1-D)
    i32x8 z8 = {0, 0, 0, 0, 0, 0, 0, 0};
    __builtin_amdgcn_tensor_load_to_lds(g0, g1, z4, z4, z8, 0);
}

// ---------------------------------------------------------------------------
// Kernel 1: per row -- obs renorm + cumtrapz (block scan), invert CDF at the
// 257 surviving knot probabilities, emit yk row and tridiagonal RHS row.
// ---------------------------------------------------------------------------
__global__ __launch_bounds__(BLK)
void k_quantiles(const float* __restrict__ obs, const float* __restrict__ tg,
                 const float* __restrict__ pg, float* __restrict__ yk_out,
                 float* __restrict__ rhs_out)
{
    __shared__ __align__(16) float sv[TLEN];   // row values, then u^2 + eps
    __shared__ __align__(16) float scn[TLEN];  // cumulative trapezoid (unnorm.)
    __shared__ float part[BLK];
    __shared__ float ykl[NK];

    const int row = blockIdx.x;
    const int tid = threadIdx.x;
    const float* o = obs + (size_t)row * TLEN;

    // --- TDM: DMA the whole row into LDS (wave 0 issues; TENSORcnt-tracked)
    if (tid < 32) {
        tdm_load_row_f32((unsigned)(unsigned long long)(void*)&sv[0], o, TLEN);
        __builtin_amdgcn_s_wait_tensorcnt(0);
    }
    __syncthreads();

    // u^2 + eps in place (each element owned by exactly one thread)
    for (int i = tid; i < TLEN; i += BLK) {
        float v = sv[i];
        sv[i] = v * v + EPSF;
    }
    __syncthreads();

    // --- blocked prefix scan of 4096 trapezoid segment areas ---
    const int base = tid * 16;
    float ps = 0.f;
    for (int k = 0; k < 16; ++k)
        ps += 0.5f * (sv[base + k] + sv[base + k + 1]) * (tg[base + k + 1] - tg[base + k]);
    part[tid] = ps;
    __syncthreads();
    for (int off = 1; off < BLK; off <<= 1) {
        float v = part[tid];
        float a = (tid >= off) ? part[tid - off] : 0.f;
        __syncthreads();
        part[tid] = v + a;
        __syncthreads();
    }
    float run = (tid > 0) ? part[tid - 1] : 0.f;
    for (int k = 0; k < 16; ++k) {
        run += 0.5f * (sv[base + k] + sv[base + k + 1]) * (tg[base + k + 1] - tg[base + k]);
        scn[base + k + 1] = run;
    }
    if (tid == 0) scn[0] = 0.f;
    __syncthreads();
    const float total = part[BLK - 1];   // == trapz(u^2, t)

    // --- invert CDF at p[16k]: first i with cdf[i] >= p  <=>  scn[i] >= p*total
    for (int k = tid; k < NK; k += BLK) {
        float target = pg[k * 16] * total;
        int lo = 0, hi = TLEN;                       // searchsorted 'left'
        while (lo < hi) {
            int mid = (lo + hi) >> 1;
            if (scn[mid] < target) lo = mid + 1; else hi = mid;
        }
        int idx = (lo > TLEN - 1) ? (TLEN - 1) : lo; // clip
        ykl[k] = tg[idx];
    }
    __syncthreads();

    const float h = pg[16] - pg[0];
    for (int k = tid; k < NK; k += BLK)
        yk_out[(size_t)row * NK + k] = ykl[k];
    for (int j = tid; j < NI; j += BLK) {            // rhs padded to 256 (last = 0)
        float r = 0.f;
        if (j < NI - 1) {
            float s0 = (ykl[j + 1] - ykl[j]) / h;
            float s1 = (ykl[j + 2] - ykl[j + 1]) / h;
            r = 6.f * (s1 - s0);
        }
        rhs_out[(size_t)row * NI + j] = r;
    }
}

// ---------------------------------------------------------------------------
// Kernel 2: build Winv (256x256).  A = h*tridiag(1,4,1), n=255; thread j does
// a Thomas solve for basis vector e_j and writes row j of Winv = (A^-1 col j)
// (A symmetric).  Row/col 255 is zero padding so m[255] == 0.
// ---------------------------------------------------------------------------
__global__ __launch_bounds__(BLK)
void k_build_winv(const float* __restrict__ pg, float* __restrict__ winv)
{
    const int j = threadIdx.x;
    const float h = pg[16] - pg[0];
    const int n = NI - 1;                 // 255 unknowns
    if (j >= n) {                         // padding row
        for (int i = 0; i < NI; ++i) winv[(size_t)j * NI + i] = 0.f;
        return;
    }
    float cp[NI];
    float dp[NI];
    cp[0] = 0.25f;
    dp[0] = ((j == 0) ? 1.f : 0.f) * 0.25f;
    for (int i = 1; i < n; ++i) {
        float m = 4.f - cp[i - 1];
        cp[i] = 1.f / m;
        dp[i] = (((i == j) ? 1.f : 0.f) - dp[i - 1]) / m;
    }
    for (int i = n - 2; i >= 0; --i)
        dp[i] = dp[i] - cp[i] * dp[i + 1];
    const float inv_h = 1.f / h;
    for (int i = 0; i < n; ++i)
        winv[(size_t)j * NI + i] = dp[i] * inv_h;
    winv[(size_t)j * NI + (NI - 1)] = 0.f;
}

// ---------------------------------------------------------------------------
// Kernel 3: M(2048x256) = RHS(2048x256) x Winv(256x256), f32 WMMA.
// One wave per 16x16 output tile, K-loop of 64 x V_WMMA_F32_16X16X4_F32.
// A 16x4 layout: lanes 0-15 hold M rows; VGPR0/1 = K pair {2*half, 2*half+1}.
// ---------------------------------------------------------------------------
__global__ __launch_bounds__(32)
void k_gemm_wmma(const float* __restrict__ rhs, const float* __restrict__ winv,
                 float* __restrict__ mout)
{
    const int lane  = threadIdx.x;
    const int half  = lane >> 4;
    const int lr    = lane & 15;
    const int tileM = blockIdx.x * 16;   // batch rows
    const int tileN = blockIdx.y * 16;   // m-vector columns

    v8f acc = {};
    for (int k = 0; k < NI; k += 4) {
        v2f a, b;
        const float* ar = rhs  + (size_t)(tileM + lr) * NI + (k + 2 * half);
        a.x = ar[0];
        a.y = ar[1];
        const float* br = winv + (size_t)(k + 2 * half) * NI + tileN + lr;
        b.x = br[0];        // B[k+2*half][n]
        b.y = br[NI];       // B[k+2*half+1][n]
        acc = __builtin_amdgcn_wmma_f32_16x16x4_f32(
            /*neg_a=*/false, a, /*neg_b=*/false, b,
            /*c_mod=*/(short)0, acc, /*reuse_a=*/false, /*reuse_b=*/false);
    }
    // C/D layout: VGPR r -> M = r (lanes 0-15) or M = r+8 (lanes 16-31), N = lr
    for (int r = 0; r < 8; ++r) {
        int mr = r + 8 * half;
        mout[(size_t)(tileM + mr) * NI + tileN + lr] = acc[r];
    }
}

// ---------------------------------------------------------------------------
// Kernel 4: per row -- traces renorm + cumtrapz (block scan), spline eval at
// each CDF value, trapezoid-integrate diff^2 * pdf -> loss[row].
// TDM row DMA is overlapped with the spline-coefficient setup.
// ---------------------------------------------------------------------------
__global__ __launch_bounds__(BLK)
void k_loss(const float* __restrict__ traces, const float* __restrict__ tg,
            const float* __restrict__ pg, const float* __restrict__ yk_in,
            const float* __restrict__ m_in, float* __restrict__ out)
{
    __shared__ __align__(16) float sv[TLEN];
    __shared__ __align__(16) float scn[TLEN];
    __shared__ __align__(16) float wbuf[TLEN];
    __shared__ float part[BLK];
    __shared__ float xk[NK];
    __shared__ float ca[NI], cb[NI], cc[NI], cd[NI];

    const int row = blockIdx.x;
    const int tid = threadIdx.x;
    const float* tr = traces + (size_t)row * TLEN;

    // --- kick off the row DMA first; coefficient setup overlaps it ---
    if (tid < 32)
        tdm_load_row_f32((unsigned)(unsigned long long)(void*)&sv[0], tr, TLEN);

    for (int k = tid; k < NK; k += BLK) xk[k] = pg[k * 16];
    __syncthreads();
    const float h = xk[1] - xk[0];

    // spline coefficients from yk row and solved m row (m_n = 0 at n=0,256)
    const float* ykr = yk_in + (size_t)row * NK;
    const float* mr  = m_in  + (size_t)row * NI;
    for (int i = tid; i < NI; i += BLK) {
        float mi  = (i == 0)      ? 0.f : mr[i - 1];     // m[i]
        float mip = (i == NI - 1) ? 0.f : mr[i];         // m[i+1]
        float a = ykr[i];
        float slope = (ykr[i + 1] - ykr[i]) / h;
        ca[i] = a;
        cb[i] = slope - h * (2.f * mi + mip) * (1.f / 6.f);
        cc[i] = 0.5f * mi;
        cd[i] = (mip - mi) / (6.f * h);
    }

    // --- wait for the DMA, then square in place ---
    if (tid < 32)
        __builtin_amdgcn_s_wait_tensorcnt(0);
    __syncthreads();
    for (int i = tid; i < TLEN; i += BLK) {
        float v = sv[i];
        sv[i] = v * v + EPSF;
    }
    __syncthreads();

    // --- blocked prefix scan (same scheme as kernel 1) ---
    const int base = tid * 16;
    float ps = 0.f;
    for (int k = 0; k < 16; ++k)
        ps += 0.5f * (sv[base + k] + sv[base + k + 1]) * (tg[base + k + 1] - tg[base + k]);
    part[tid] = ps;
    __syncthreads();
    for (int off = 1; off < BLK; off <<= 1) {
        float v = part[tid];
        float a = (tid >= off) ? part[tid - off] : 0.f;
        __syncthreads();
        part[tid] = v + a;
        __syncthreads();
    }
    float run = (tid > 0) ? part[tid - 1] : 0.f;
    for (int k = 0; k < 16; ++k) {
        run += 0.5f * (sv[base + k] + sv[base + k + 1]) * (tg[base + k + 1] - tg[base + k]);
        scn[base + k + 1] = run;
    }
    if (tid == 0) scn[0] = 0.f;
    __syncthreads();
    const float total = part[BLK - 1];
    const float inv_total = 1.f / total;

    // --- evaluate spline at cdf, form diff^2 * pdf ---
    for (int j = tid; j < TLEN; j += BLK) {
        float q = scn[j] * inv_total;
        int lo = 0, hi = NK;                    // searchsorted(xk, q, 'left')
        while (lo < hi) {
            int mid = (lo + hi) >> 1;
            if (xk[mid] < q) lo = mid + 1; else hi = mid;
        }
        int idx = lo - 1;
        idx = idx < 0 ? 0 : (idx > NI - 1 ? NI - 1 : idx);
        float f = q - xk[idx];
        float s = ca[idx] + f * (cb[idx] + f * (cc[idx] + cd[idx] * f));
        float diff = tg[j] - s;
        wbuf[j] = diff * diff * (sv[j] * inv_total);
    }
    __syncthreads();

    // --- trapezoid reduce ---
    float acc = 0.f;
    for (int k = 0; k < 16; ++k)
        acc += 0.5f * (wbuf[base + k] + wbuf[base + k + 1]) * (tg[base + k + 1] - tg[base + k]);
    part[tid] = acc;
    __syncthreads();
    for (int off = BLK >> 1; off > 0; off >>= 1) {
        if (tid < off) part[tid] += part[tid + off];
        __syncthreads();
    }
    if (tid == 0) out[row] = part[0];
}

// ---------------------------------------------------------------------------
// Host-side launcher
// ---------------------------------------------------------------------------
extern "C" void kernel_launch(void* const* d_in, const int* in_sizes, int n_in,
                              void* d_out, int out_size, void* d_ws, size_t ws_size,
                              hipStream_t stream)
{
    const float* traces = (const float*)d_in[0];   // (B, T)
    const float* obs    = (const float*)d_in[1];   // (B, T)
    const float* tg     = (const float*)d_in[2];   // (T,)
    const float* pg     = (const float*)d_in[3];   // (P,)
    (void)d_in[4];                                 // down == 16 (structural)

    const int B = in_sizes[0] / TLEN;              // 2048

    // workspace layout (floats)
    float* ws   = (float*)d_ws;
    float* yk   = ws;                                // B * 257
    float* rhs  = yk   + (size_t)B * NK;             // B * 256
    float* winv = rhs  + (size_t)B * NI;             // 256 * 256
    float* mout = winv + (size_t)NI * NI;            // B * 256
    float* out  = (float*)d_out;                     // (B,)

    k_quantiles<<<B, BLK, 0, stream>>>(obs, tg, pg, yk, rhs);
    k_build_winv<<<1, BLK, 0, stream>>>(pg, winv);
    dim3 ggrid(B / 16, NI / 16);                     // (128, 16) 16x16 tiles
    k_gemm_wmma<<<ggrid, 32, 0, stream>>>(rhs, winv, mout);
    k_loss<<<B, BLK, 0, stream>>>(traces, tg, pg, yk, mout, out);
}